// ProsGraphNet_87591563035163
// MI455X (gfx1250) — compile-verified
//
#include <hip/hip_runtime.h>
#include <math.h>

#define NN 131072          // total nodes per side
#define SS 16384           // nodes per graph
#define BB 8               // graphs
#define DEG 16
#define GSCALE (1.0f/17.0f)

typedef float f4  __attribute__((ext_vector_type(4)));
typedef float v2f __attribute__((ext_vector_type(2)));
typedef float v8f __attribute__((ext_vector_type(8)));

__device__ __forceinline__ f4 relu4(f4 v) {
    f4 r;
    #pragma unroll
    for (int i = 0; i < 4; ++i) r[i] = fmaxf(v[i], 0.0f);
    return r;
}

// ---------------- GCN layer 1 (fused pre-linear aggregate): x[N,3] -> h1[N,32] ----------------
__global__ __launch_bounds__(256) void gcn1_kernel(const float* __restrict__ x,
                                                   const int* __restrict__ src,
                                                   const float* __restrict__ W1,
                                                   const float* __restrict__ b1,
                                                   float* __restrict__ h1) {
    __shared__ float Wl[96];
    __shared__ float bl[32];
    int tid = threadIdx.x;
    if (tid < 96) Wl[tid] = W1[tid];
    if (tid < 32) bl[tid] = b1[tid];
    __syncthreads();
    int n = blockIdx.x * 256 + tid;
    float sx = x[n * 3 + 0], sy = x[n * 3 + 1], sz = x[n * 3 + 2];
    const int* sp = src + n * DEG;
    for (int k = 0; k < DEG; ++k) {
        int j = sp[k];
        sx += x[j * 3 + 0]; sy += x[j * 3 + 1]; sz += x[j * 3 + 2];
    }
    const f4* W4 = (const f4*)Wl;      // [3][8] rows of f4
    const f4* b4 = (const f4*)bl;
    f4* out = (f4*)(h1 + (size_t)n * 32);
    #pragma unroll
    for (int i = 0; i < 8; ++i) {
        f4 acc = sx * W4[i] + sy * W4[8 + i] + sz * W4[16 + i];
        out[i] = relu4(GSCALE * acc + b4[i]);
    }
}

// ---------------- GCN layer 2: h1[N,32] -> h2[N,24] ----------------
__global__ __launch_bounds__(256) void gcn2_kernel(const float* __restrict__ h1,
                                                   const int* __restrict__ src,
                                                   const float* __restrict__ W2,
                                                   const float* __restrict__ b2,
                                                   float* __restrict__ h2) {
    __shared__ float Wl[768];
    __shared__ float bl[24];
    int tid = threadIdx.x;
    for (int i = tid; i < 768; i += 256) Wl[i] = W2[i];
    if (tid < 24) bl[tid] = b2[tid];
    __syncthreads();
    int n = blockIdx.x * 256 + tid;
    const f4* hv = (const f4*)h1;      // row stride 8 (32 floats)
    f4 s[8];
    #pragma unroll
    for (int i = 0; i < 8; ++i) s[i] = hv[(size_t)n * 8 + i];
    const int* sp = src + n * DEG;
    for (int k = 0; k < DEG; ++k) {
        size_t j = (size_t)sp[k] * 8;
        #pragma unroll
        for (int i = 0; i < 8; ++i) s[i] += hv[j + i];
    }
    f4 acc[6];
    #pragma unroll
    for (int i = 0; i < 6; ++i) acc[i] = (f4){};
    const f4* W4 = (const f4*)Wl;      // row c: 6 f4 at c*6
    #pragma unroll
    for (int c = 0; c < 32; ++c) {
        float sc = s[c >> 2][c & 3];
        #pragma unroll
        for (int i = 0; i < 6; ++i) acc[i] += sc * W4[c * 6 + i];
    }
    const f4* b4 = (const f4*)bl;
    f4* out = (f4*)(h2 + (size_t)n * 24);
    #pragma unroll
    for (int i = 0; i < 6; ++i) out[i] = relu4(GSCALE * acc[i] + b4[i]);
}

// ---------------- GCN layer 3 + L2 norm: h2[N,24] -> f[N,16] ----------------
__global__ __launch_bounds__(256) void gcn3_kernel(const float* __restrict__ h2,
                                                   const int* __restrict__ src,
                                                   const float* __restrict__ W3,
                                                   const float* __restrict__ b3,
                                                   float* __restrict__ f) {
    __shared__ float Wl[384];
    __shared__ float bl[16];
    int tid = threadIdx.x;
    for (int i = tid; i < 384; i += 256) Wl[i] = W3[i];
    if (tid < 16) bl[tid] = b3[tid];
    __syncthreads();
    int n = blockIdx.x * 256 + tid;
    const f4* hv = (const f4*)h2;      // row stride 6 (24 floats)
    f4 s[6];
    #pragma unroll
    for (int i = 0; i < 6; ++i) s[i] = hv[(size_t)n * 6 + i];
    const int* sp = src + n * DEG;
    for (int k = 0; k < DEG; ++k) {
        size_t j = (size_t)sp[k] * 6;
        #pragma unroll
        for (int i = 0; i < 6; ++i) s[i] += hv[j + i];
    }
    f4 acc[4];
    #pragma unroll
    for (int i = 0; i < 4; ++i) acc[i] = (f4){};
    const f4* W4 = (const f4*)Wl;      // row c: 4 f4 at c*4
    #pragma unroll
    for (int c = 0; c < 24; ++c) {
        float sc = s[c >> 2][c & 3];
        #pragma unroll
        for (int i = 0; i < 4; ++i) acc[i] += sc * W4[c * 4 + i];
    }
    const f4* b4 = (const f4*)bl;
    f4 v[4];
    float ss = 1e-6f;
    #pragma unroll
    for (int i = 0; i < 4; ++i) {
        v[i] = GSCALE * acc[i] + b4[i];
        #pragma unroll
        for (int e = 0; e < 4; ++e) ss += v[i][e] * v[i][e];
    }
    float inv = 1.0f / sqrtf(ss);
    f4* out = (f4*)(f + (size_t)n * 16);
    #pragma unroll
    for (int i = 0; i < 4; ++i) out[i] = v[i] * inv;
}

// ---------------- Correlation: per graph, C = fa^T @ fb (K=16384) via V_WMMA_F32_16X16X4_F32 ----
// A (16x4 f32): lane l<16 -> M=l, v0=K0, v1=K1 ; lane l+16 -> v0=K2, v1=K3
// B (4x16 f32): lane l<16 -> N=l, v0=K0, v1=K1 ; lane l+16 -> v0=K2, v1=K3
// C/D (16x16): vgpr r, lanes 0-15 -> (M=r, N=lane); lanes 16-31 -> (M=8+r, N=lane-16)
__global__ __launch_bounds__(256) void corr_kernel(const float* __restrict__ fA,
                                                   const float* __restrict__ fB,
                                                   float* __restrict__ corr) {
    __shared__ float red[8 * 256];
    __shared__ float vals[256];
    __shared__ float cn[16];
    int tid  = threadIdx.x;
    int g    = blockIdx.x;
    int wave = tid >> 5, lane = tid & 31;
    int c = lane & 15, half = lane >> 4;
    const float* fa = fA + (size_t)g * SS * 16;
    const float* fb = fB + (size_t)g * SS * 16;
    int base = wave * 2048 + half * 2;   // first row this lane reads per 4-row step
    v8f acc0 = {}, acc1 = {};
    for (int t = 0; t < 512; t += 2) {
        int n = base + t * 4;
        v2f a, b;
        a[0] = fa[(size_t)n * 16 + c];       a[1] = fa[(size_t)(n + 1) * 16 + c];
        b[0] = fb[(size_t)n * 16 + c];       b[1] = fb[(size_t)(n + 1) * 16 + c];
        acc0 = __builtin_amdgcn_wmma_f32_16x16x4_f32(false, a, false, b, (short)0, acc0, false, false);
        int n2 = n + 4;
        v2f a2, b2;
        a2[0] = fa[(size_t)n2 * 16 + c];     a2[1] = fa[(size_t)(n2 + 1) * 16 + c];
        b2[0] = fb[(size_t)n2 * 16 + c];     b2[1] = fb[(size_t)(n2 + 1) * 16 + c];
        acc1 = __builtin_amdgcn_wmma_f32_16x16x4_f32(false, a2, false, b2, (short)0, acc1, false, false);
    }
    v8f acc = acc0 + acc1;
    #pragma unroll
    for (int r = 0; r < 8; ++r)
        red[wave * 256 + (r + 8 * half) * 16 + c] = acc[r];
    __syncthreads();
    float v = 0.f;
    #pragma unroll
    for (int w = 0; w < 8; ++w) v += red[w * 256 + tid];
    v = fmaxf(v, 0.f);                    // relu
    vals[tid] = v;
    __syncthreads();
    if (tid < 16) {                        // L2 norm over axis=1 (channel c) per column d
        float ss = 1e-6f;
        #pragma unroll
        for (int cc = 0; cc < 16; ++cc) { float t2 = vals[cc * 16 + tid]; ss += t2 * t2; }
        cn[tid] = 1.0f / sqrtf(ss);
    }
    __syncthreads();
    corr[g * 256 + tid] = vals[tid] * cn[tid & 15];
}

// ---------------- Head: conv(4x4 VALID)->BN->relu->conv(3x3 pad1 on 1x1)->BN->relu->linear ----
__global__ __launch_bounds__(256) void head_kernel(const float* __restrict__ corr,
                                                   const float* __restrict__ c1w, const float* __restrict__ c1b,
                                                   const float* __restrict__ g1,  const float* __restrict__ be1,
                                                   const float* __restrict__ c2w, const float* __restrict__ c2b,
                                                   const float* __restrict__ g2,  const float* __restrict__ be2,
                                                   const float* __restrict__ lw,  const float* __restrict__ lb,
                                                   float* __restrict__ out) {
    __shared__ float xin[8 * 256];
    __shared__ float y1[8 * 128];
    __shared__ float y2[8 * 64];
    int tid = threadIdx.x;
    for (int i = tid; i < 2048; i += 256) xin[i] = corr[i];
    __syncthreads();
    // conv1: 4x4 VALID on 4x4 == full dot over 16*4*4=256
    for (int t = tid; t < 1024; t += 256) {
        int b = t >> 7, o = t & 127;
        float acc = c1b[o];
        const float* w  = c1w + o * 256;
        const float* xi = xin + b * 256;
        for (int i = 0; i < 256; ++i) acc += xi[i] * w[i];
        y1[b * 128 + o] = acc;
    }
    __syncthreads();
    if (tid < 128) {   // BN over batch (N,H,W)=(8,1,1) + relu
        float m = 0.f;
        for (int b = 0; b < 8; ++b) m += y1[b * 128 + tid];
        m *= 0.125f;
        float v = 0.f;
        for (int b = 0; b < 8; ++b) { float d = y1[b * 128 + tid] - m; v += d * d; }
        v *= 0.125f;
        float inv = 1.0f / sqrtf(v + 1e-5f);
        float gg = g1[tid], bb = be1[tid];
        for (int b = 0; b < 8; ++b)
            y1[b * 128 + tid] = fmaxf((y1[b * 128 + tid] - m) * inv * gg + bb, 0.f);
    }
    __syncthreads();
    // conv2: 3x3 pad 1 on 1x1 input -> only center tap (1,1) contributes
    for (int t = tid; t < 512; t += 256) {
        int b = t >> 6, o = t & 63;
        float acc = c2b[o];
        const float* yi = y1 + b * 128;
        for (int i = 0; i < 128; ++i) acc += yi[i] * c2w[(o * 128 + i) * 9 + 4];
        y2[b * 64 + o] = acc;
    }
    __syncthreads();
    if (tid < 64) {    // BN2 + relu
        float m = 0.f;
        for (int b = 0; b < 8; ++b) m += y2[b * 64 + tid];
        m *= 0.125f;
        float v = 0.f;
        for (int b = 0; b < 8; ++b) { float d = y2[b * 64 + tid] - m; v += d * d; }
        v *= 0.125f;
        float inv = 1.0f / sqrtf(v + 1e-5f);
        float gg = g2[tid], bb = be2[tid];
        for (int b = 0; b < 8; ++b)
            y2[b * 64 + tid] = fmaxf((y2[b * 64 + tid] - m) * inv * gg + bb, 0.f);
    }
    __syncthreads();
    if (tid < 48) {
        int b = tid / 6, j = tid % 6;
        float acc = lb[j];
        const float* yi = y2 + b * 64;
        for (int i = 0; i < 64; ++i) acc += yi[i] * lw[j * 64 + i];
        const float ident[6] = {1.f, 0.f, 0.f, 0.f, 1.f, 0.f};
        out[b * 6 + j] = 0.1f * acc + ident[j];
    }
}

extern "C" void kernel_launch(void* const* d_in, const int* in_sizes, int n_in,
                              void* d_out, int out_size, void* d_ws, size_t ws_size,
                              hipStream_t stream) {
    const float* x_A  = (const float*)d_in[0];
    const float* x_B  = (const float*)d_in[1];
    const int*   eA   = (const int*)  d_in[2];   // [2,E], row 0 = src (node n's edges at n*16)
    const int*   eB   = (const int*)  d_in[3];
    const float* W1   = (const float*)d_in[4];
    const float* b1   = (const float*)d_in[5];
    const float* W2   = (const float*)d_in[6];
    const float* b2   = (const float*)d_in[7];
    const float* W3   = (const float*)d_in[8];
    const float* b3   = (const float*)d_in[9];
    const float* c1w  = (const float*)d_in[10];
    const float* c1b  = (const float*)d_in[11];
    const float* g1   = (const float*)d_in[12];
    const float* be1  = (const float*)d_in[13];
    const float* c2w  = (const float*)d_in[14];
    const float* c2b  = (const float*)d_in[15];
    const float* g2   = (const float*)d_in[16];
    const float* be2  = (const float*)d_in[17];
    const float* lw   = (const float*)d_in[18];
    const float* lb   = (const float*)d_in[19];
    float* out = (float*)d_out;

    float* wsf  = (float*)d_ws;
    float* fA   = wsf;                         // N*16
    float* fB   = wsf + (size_t)NN * 16;       // N*16
    float* h1   = wsf + (size_t)NN * 32;       // N*32
    float* h2   = wsf + (size_t)NN * 64;       // N*24
    float* corr = wsf + (size_t)NN * 88;       // 8*16*16

    dim3 gB(NN / 256), blk(256);
    // side A
    gcn1_kernel<<<gB, blk, 0, stream>>>(x_A, eA, W1, b1, h1);
    gcn2_kernel<<<gB, blk, 0, stream>>>(h1, eA, W2, b2, h2);
    gcn3_kernel<<<gB, blk, 0, stream>>>(h2, eA, W3, b3, fA);
    // side B (reuse h1/h2 scratch)
    gcn1_kernel<<<gB, blk, 0, stream>>>(x_B, eB, W1, b1, h1);
    gcn2_kernel<<<gB, blk, 0, stream>>>(h1, eB, W2, b2, h2);
    gcn3_kernel<<<gB, blk, 0, stream>>>(h2, eB, W3, b3, fB);
    // per-graph correlation (WMMA f32 16x16x4, K=16384) + relu + column L2 norm
    corr_kernel<<<dim3(BB), blk, 0, stream>>>(fA, fB, corr);
    // tiny conv/BN/linear head -> theta [8,2,3]
    head_kernel<<<dim3(1), blk, 0, stream>>>(corr, c1w, c1b, g1, be1,
                                             c2w, c2b, g2, be2, lw, lb, out);
}